// DisplaceChannel_7499012899665
// MI455X (gfx1250) — compile-verified
//
#include <hip/hip_runtime.h>
#include <stdint.h>

// Problem constants (from the reference): B=32, C=384 (48 pos x 8 ch), H=W=64, K=3, sigma=0.5
#define HH 64
#define WW 64
#define CC 384
#define BB 32

// One workgroup per (b, c) plane. 256 threads = 8 wave32.
// Stage the whole 64x64 f32 plane (16 KB) in LDS via CDNA5 async global->LDS
// with non-temporal hint (streaming data, zero reuse across WGs; in+out > 192MB L2),
// then fused displace + separable 3x3 Gaussian: one HBM read + one NT HBM write per element.
__global__ __launch_bounds__(256) void displace_gauss_kernel(
    const float* __restrict__ inp,
    const float* __restrict__ offset,
    float* __restrict__ out)
{
    __shared__ float tile[HH * WW];   // 16 KB

    const int bc  = blockIdx.x;       // b*CC + c
    const int c   = bc % CC;
    const int p   = c >> 3;           // position index (8 channels per position)
    const int tid = threadIdx.x;

    const float* src = inp + (size_t)bc * (HH * WW);
    float*       dst = out + (size_t)bc * (HH * WW);

    // ---- CDNA5 async copy: 64x64 plane -> LDS (4 x b128 per lane, coalesced, NT) ----
    const uint32_t lds_base = (uint32_t)(uintptr_t)(&tile[0]);
    #pragma unroll
    for (int i = 0; i < 4; ++i) {
        const int v4 = tid + i * 256;                       // float4 index 0..1023
        uint64_t ga = (uint64_t)(uintptr_t)src + (uint64_t)v4 * 16u;
        uint32_t la = lds_base + (uint32_t)v4 * 16u;
        asm volatile("global_load_async_to_lds_b128 %0, %1, off th:TH_LOAD_NT"
                     :: "v"(la), "v"(ga) : "memory");
    }

    // ---- per-position integer shift + separable, normalized Gaussian weights ----
    // (uniform per block; 6 exps, negligible)
    const float offx = offset[2 * p + 0];
    const float offy = offset[2 * p + 1];
    const float rx = rintf(offx);                 // matches jnp.round (never exactly .5 here)
    const float ry = rintf(offy);
    const int   ox = (int)rx;
    const int   oy = (int)ry;
    const float fx = offx - rx;
    const float fy = offy - ry;

    float wx[3], wy[3];
    float sx = 0.f, sy = 0.f;
    #pragma unroll
    for (int k = 0; k < 3; ++k) {
        const float dxv = (float)(k - 1) + fx;
        const float dyv = (float)(k - 1) + fy;
        wx[k] = expf(-2.0f * dxv * dxv);  sx += wx[k];   // 2*sigma^2 = 0.5
        wy[k] = expf(-2.0f * dyv * dyv);  sy += wy[k];
    }
    const float isx = 1.0f / sx, isy = 1.0f / sy;
    #pragma unroll
    for (int k = 0; k < 3; ++k) { wx[k] *= isx; wy[k] *= isy; }

    // ---- per-thread column strip: x fixed, 16 consecutive y ----
    const int x  = tid & (WW - 1);
    const int y0 = (tid >> 6) * 16;

    // Horizontal taps: source column + combined (conv-padding AND displacement) mask
    int   xs[3];
    float mwx[3];
    #pragma unroll
    for (int k = 0; k < 3; ++k) {
        const int  xc = x + k - 1;       // conv column
        const int  xv = xc - ox;         // shifted source column
        const bool v  = ((unsigned)xc < WW) && ((unsigned)xv < WW);
        xs[k]  = v ? xv : 0;
        mwx[k] = v ? wx[k] : 0.0f;
    }

    asm volatile("s_wait_asynccnt 0" ::: "memory");  // my wave's async copies done
    __syncthreads();                                  // everyone's copies visible

    // Horizontal 3-tap sum for conv row yy (mask folded in; 3 LDS reads)
    auto hrow = [&](int yy) -> float {
        const int  yv = yy - oy;
        const bool v  = ((unsigned)yy < HH) && ((unsigned)yv < HH);
        const int  base = (v ? yv : 0) * WW;
        const float h = mwx[0] * tile[base + xs[0]]
                      + mwx[1] * tile[base + xs[1]]
                      + mwx[2] * tile[base + xs[2]];
        return v ? h : 0.0f;
    };

    // Rolling vertical combine: 3 new LDS reads + 3 FMA per output, NT store
    float hm = hrow(y0 - 1);
    float hc = hrow(y0);
    #pragma unroll
    for (int i = 0; i < 16; ++i) {
        const int   y  = y0 + i;
        const float hp = hrow(y + 1);
        const float o  = wy[0] * hm + wy[1] * hc + wy[2] * hp;
        __builtin_nontemporal_store(o, &dst[y * WW + x]);
        hm = hc; hc = hp;
    }
}

extern "C" void kernel_launch(void* const* d_in, const int* in_sizes, int n_in,
                              void* d_out, int out_size, void* d_ws, size_t ws_size,
                              hipStream_t stream) {
    const float* inp    = (const float*)d_in[0];   // (B, C, H, W) f32
    const float* offset = (const float*)d_in[1];   // (48, 2) f32
    float*       out    = (float*)d_out;           // (B, C, H, W) f32

    displace_gauss_kernel<<<dim3(BB * CC), dim3(256), 0, stream>>>(inp, offset, out);
}